// T1DPatient_11373073400314
// MI455X (gfx1250) — compile-verified
//
#include <hip/hip_runtime.h>
#include <hip/hip_bf16.h>
#include <stdint.h>

#define TILE 256
#define NX 13
#define NP 30

// --- CDNA5 async Global<->LDS helpers (gfx1250, ASYNCcnt-tracked) -----------
// Streaming workload (480 MB, zero reuse vs 192 MB L2) -> non-temporal hints.

__device__ __forceinline__ uint32_t lds_lo(const void* p) {
  // generic pointer to __shared__ carries the LDS byte offset in its low 32
  // bits (LDS aperture is 4GB-aligned)
  return (uint32_t)(uintptr_t)p;
}

__device__ __forceinline__ void async_ld_b128(uint32_t lds, const void* g) {
  asm volatile("global_load_async_to_lds_b128 %0, %1, off th:TH_LOAD_NT"
               :: "v"(lds), "v"((uint64_t)(uintptr_t)g) : "memory");
}
__device__ __forceinline__ void async_ld_b32(uint32_t lds, const void* g) {
  asm volatile("global_load_async_to_lds_b32 %0, %1, off th:TH_LOAD_NT"
               :: "v"(lds), "v"((uint64_t)(uintptr_t)g) : "memory");
}
__device__ __forceinline__ void async_st_b128(void* g, uint32_t lds) {
  asm volatile("global_store_async_from_lds_b128 %0, %1, off th:TH_STORE_NT"
               :: "v"((uint64_t)(uintptr_t)g), "v"(lds) : "memory");
}
__device__ __forceinline__ void async_st_b32(void* g, uint32_t lds) {
  asm volatile("global_store_async_from_lds_b32 %0, %1, off th:TH_STORE_NT"
               :: "v"((uint64_t)(uintptr_t)g), "v"(lds) : "memory");
}
__device__ __forceinline__ void wait_async0() {
#if __has_builtin(__builtin_amdgcn_s_wait_asynccnt)
  __builtin_amdgcn_s_wait_asynccnt(0);
#else
  asm volatile("s_wait_asynccnt 0" ::: "memory");
#endif
}

// ---------------------------------------------------------------------------

__global__ __launch_bounds__(TILE) void t1d_dxdt_kernel(
    const float* __restrict__ xg, const float* __restrict__ pg,
    const float* __restrict__ CHOg, const float* __restrict__ insg,
    const float* __restrict__ lqg, const float* __restrict__ lfg,
    float* __restrict__ outg, int B) {
  __shared__ __align__(16) float xs[TILE * NX];   // x tile in, reused as out tile
  __shared__ __align__(16) float ps[TILE * NP];   // params tile

  const int tid  = threadIdx.x;
  const int base = blockIdx.x * TILE;
  const int n    = (B - base < TILE) ? (B - base) : TILE;
  const bool active = tid < n;

  // ---- stage x tile FIRST (coalesced async b128 Global->LDS, ASYNCcnt) ----
  {
    const float* gx = xg + (size_t)base * NX;
    const int tot = n * NX, chunks = tot >> 2, tail = tot & 3;
    for (int c = tid; c < chunks; c += TILE)
      async_ld_b128(lds_lo(xs) + (uint32_t)c * 16u, gx + c * 4);
    if (tid < tail)
      async_ld_b32(lds_lo(xs) + (uint32_t)(chunks * 4 + tid) * 4u,
                   gx + chunks * 4 + tid);
  }
  // ---- stage params tile ----
  {
    const float* gp = pg + (size_t)base * NP;
    const int tot = n * NP, chunks = tot >> 2, tail = tot & 3;
    for (int c = tid; c < chunks; c += TILE)
      async_ld_b128(lds_lo(ps) + (uint32_t)c * 16u, gp + c * 4);
    if (tid < tail)
      async_ld_b32(lds_lo(ps) + (uint32_t)(chunks * 4 + tid) * 4u,
                   gp + chunks * 4 + tid);
  }

  // ---- per-element scalar loads AFTER DMA issue (LOADcnt path, unit-stride;
  //      their latency + any compiler waits overlap the outstanding async DMA;
  //      the asm "memory" clobbers keep them from being hoisted above it) ----
  float cho = 0.f, insu = 0.f, lq = 0.f, lf = 0.f;
  if (active) {
    const int i = base + tid;
    cho  = __builtin_nontemporal_load(CHOg + i);
    insu = __builtin_nontemporal_load(insg + i);
    lq   = __builtin_nontemporal_load(lqg + i);
    lf   = __builtin_nontemporal_load(lfg + i);
  }

  wait_async0();       // this wave's async LDS writes landed
  __syncthreads();     // all waves' tiles visible

  float dr[NX];

  if (active) {
    // LDS reads: stride 13 / stride 30 across lanes -> bank-conflict-free (64 banks)
    float xr[NX];
#pragma unroll
    for (int j = 0; j < NX; ++j) xr[j] = xs[tid * NX + j];
    float pr[NP];
#pragma unroll
    for (int j = 0; j < NP; ++j) pr[j] = ps[tid * NP + j];

    const float kmax = pr[0],  kmin = pr[1],  b   = pr[2],  d_  = pr[3];
    const float kabs = pr[4],  f    = pr[5],  BW  = pr[6],  kp1 = pr[7];
    const float kp2  = pr[8],  kp3  = pr[9],  Fsnc= pr[10], ke1 = pr[11];
    const float ke2  = pr[12], k1   = pr[13], k2  = pr[14], Vm0 = pr[15];
    const float Vmx  = pr[16], Km0  = pr[17], m1  = pr[18], m2  = pr[19];
    const float m4   = pr[20], m30  = pr[21], ka1 = pr[22], ka2 = pr[23];
    const float kd   = pr[24], Vi   = pr[25], p2u = pr[26], Ib  = pr[27];
    const float ki   = pr[28], ksc  = pr[29];

    const float x0 = xr[0], x1 = xr[1], x2 = xr[2], x3 = xr[3], x4 = xr[4];
    const float x5 = xr[5], x6 = xr[6], x7 = xr[7], x8 = xr[8], x9 = xr[9];
    const float x10 = xr[10], x11 = xr[11], x12 = xr[12];

    const float dmg     = cho * 1000.0f;
    const float insulin = insu * 6000.0f / BW;

    const float qsto = x0 + x1;
    const float Dbar = lq + lf;
    const bool  has_food  = Dbar > 0.0f;
    const float Dbar_safe = has_food ? Dbar : 1.0f;
    const float aa = 2.5f / ((1.0f - b) * Dbar_safe);
    const float cc = 2.5f / (d_ * Dbar_safe);
    const float kgut_eat = kmin + (kmax - kmin) * 0.5f *
        (tanhf(aa * (qsto - b * Dbar)) - tanhf(cc * (qsto - d_ * Dbar)) + 2.0f);
    const float kgut = has_food ? kgut_eat : kmax;

    dr[0] = -kmax * x0 + dmg;
    dr[1] =  kmax * x0 - x1 * kgut;
    dr[2] =  kgut * x1 - kabs * x2;

    const float Rat  = f * kabs * x2 / BW;
    const float EGPt = kp1 - kp2 * x3 - kp3 * x8;
    const float Et   = (x3 > ke2) ? ke1 * (x3 - ke2) : 0.0f;
    dr[3] = (x3 >= 0.0f)
              ? (fmaxf(EGPt, 0.0f) + Rat - Fsnc - Et - k1 * x3 + k2 * x4) : 0.0f;

    const float Vmt  = Vm0 + Vmx * x6;
    const float Uidt = Vmt * x4 / (Km0 + x4);
    dr[4] = (x4 >= 0.0f) ? (-Uidt + k1 * x3 - k2 * x4) : 0.0f;

    dr[5] = (x5 >= 0.0f)
              ? (-(m2 + m4) * x5 + m1 * x9 + ka1 * x10 + ka2 * x11) : 0.0f;

    const float It = x5 / Vi;
    dr[6]  = -p2u * x6 + p2u * (It - Ib);
    dr[7]  = -ki * (x7 - It);
    dr[8]  = -ki * (x8 - x7);
    dr[9]  = (x9  >= 0.0f) ? (-(m1 + m30) * x9 + m2 * x5)        : 0.0f;
    dr[10] = (x10 >= 0.0f) ? (insulin - (ka1 + kd) * x10)        : 0.0f;
    dr[11] = (x11 >= 0.0f) ? (kd * x10 - ka2 * x11)              : 0.0f;
    dr[12] = (x12 >= 0.0f) ? (-ksc * x12 + ksc * x3)             : 0.0f;
  }

  __syncthreads();   // everyone done reading xs before we overwrite it
  if (active) {
#pragma unroll
    for (int j = 0; j < NX; ++j) xs[tid * NX + j] = dr[j];
  }
  __syncthreads();   // out tile complete in LDS (dscnt waited by barrier)

  // ---- stream out tile (coalesced async b128 LDS->Global, NT) ----
  {
    float* go = outg + (size_t)base * NX;
    const int tot = n * NX, chunks = tot >> 2, tail = tot & 3;
    for (int c = tid; c < chunks; c += TILE)
      async_st_b128(go + c * 4, lds_lo(xs) + (uint32_t)c * 16u);
    if (tid < tail)
      async_st_b32(go + chunks * 4 + tid,
                   lds_lo(xs) + (uint32_t)(chunks * 4 + tid) * 4u);
  }
  wait_async0();  // (S_ENDPGM also implies wait-idle; explicit for clarity)
}

extern "C" void kernel_launch(void* const* d_in, const int* in_sizes, int n_in,
                              void* d_out, int out_size, void* d_ws, size_t ws_size,
                              hipStream_t stream) {
  const float* x    = (const float*)d_in[0];
  const float* prm  = (const float*)d_in[1];
  const float* CHO  = (const float*)d_in[2];
  const float* ins  = (const float*)d_in[3];
  const float* lq   = (const float*)d_in[4];
  const float* lf   = (const float*)d_in[5];
  float* out        = (float*)d_out;

  const int B = in_sizes[2];                 // CHO is (B,)
  const int grid = (B + TILE - 1) / TILE;
  hipLaunchKernelGGL(t1d_dxdt_kernel, dim3(grid), dim3(TILE), 0, stream,
                     x, prm, CHO, ins, lq, lf, out, B);
}